// Q_Fun_67997922231101
// MI455X (gfx1250) — compile-verified
//
#include <hip/hip_runtime.h>

#define HID 64
#define T_LAYERS 4

typedef __attribute__((ext_vector_type(2))) float v2f;
typedef __attribute__((ext_vector_type(8))) float v8f;

// ---------------------------------------------------------------------------
// One pass over edges: in-degree histogram (by dst) + deg_attr (by src).
// Both are layer-invariant, built once per launch.
// ---------------------------------------------------------------------------
__global__ void edge_stats_kernel(const int* __restrict__ src,
                                  const int* __restrict__ dst,
                                  const float* __restrict__ eattr,
                                  int* __restrict__ hist,
                                  float* __restrict__ dega, int E) {
    int e = blockIdx.x * blockDim.x + threadIdx.x;
    if (e < E) {
        atomicAdd(&hist[dst[e]], 1);
        atomicAdd(&dega[src[e]], eattr[e]);
    }
}

// ---------------------------------------------------------------------------
// Single-block exclusive scan of hist[N] -> rowptr[N] (and cursor copy).
// 1024 threads, Hillis-Steele per 1024-chunk with running carry.
// ---------------------------------------------------------------------------
__global__ __launch_bounds__(1024) void scan_kernel(const int* __restrict__ hist,
                                                    int* __restrict__ rowptr,
                                                    int* __restrict__ cursor, int N) {
    __shared__ int buf[1024];
    __shared__ int carry;
    const int tid = threadIdx.x;
    if (tid == 0) carry = 0;
    __syncthreads();
    for (int base = 0; base < N; base += 1024) {
        const int i = base + tid;
        const int v = (i < N) ? hist[i] : 0;
        buf[tid] = v;
        __syncthreads();
        #pragma unroll
        for (int off = 1; off < 1024; off <<= 1) {
            int t = (tid >= off) ? buf[tid - off] : 0;
            __syncthreads();
            buf[tid] += t;
            __syncthreads();
        }
        const int excl = buf[tid] - v + carry;
        if (i < N) { rowptr[i] = excl; cursor[i] = excl; }
        __syncthreads();                     // everyone has read carry
        if (tid == 1023) carry += buf[1023]; // chunk total
        __syncthreads();
    }
}

// ---------------------------------------------------------------------------
// Fill CSR column array: colsrc[slot in dst's row] = src  (once per launch)
// ---------------------------------------------------------------------------
__global__ void csr_fill_kernel(const int* __restrict__ src,
                                const int* __restrict__ dst,
                                int* __restrict__ cursor,
                                int* __restrict__ colsrc, int E) {
    int e = blockIdx.x * blockDim.x + threadIdx.x;
    if (e < E) {
        int p = atomicAdd(&cursor[dst[e]], 1);
        colsrc[p] = src[e];
    }
}

// ---------------------------------------------------------------------------
// aggr[n] = sum over in-neighbors s of x[s]   -- atomic-free CSR gather.
// One wave per node; lane covers 2 columns; neighbor indices loaded once
// per 32 edges and broadcast via shfl.
// ---------------------------------------------------------------------------
__global__ void gather_aggr_kernel(const int* __restrict__ rowptr,
                                   const int* __restrict__ hist,
                                   const int* __restrict__ colsrc,
                                   const float* __restrict__ x,
                                   float* __restrict__ aggr, int N) {
    const int wave   = (blockIdx.x * blockDim.x + threadIdx.x) >> 5;
    const int lane   = threadIdx.x & 31;
    const int nwaves = (gridDim.x * blockDim.x) >> 5;
    for (int n = wave; n < N; n += nwaves) {
        const int s0  = rowptr[n];
        const int deg = hist[n];
        float ax = 0.f, ay = 0.f;
        for (int jb = 0; jb < deg; jb += 32) {
            const int rem = deg - jb;
            const int cnt = rem < 32 ? rem : 32;
            const int myidx = (lane < cnt) ? colsrc[s0 + jb + lane] : 0;
            for (int t = 0; t < cnt; ++t) {
                const int s = __shfl(myidx, t);
                const float2 xv = *(const float2*)(x + (size_t)s * HID + lane * 2);
                ax += xv.x;
                ay += xv.y;
            }
        }
        float2 o; o.x = ax; o.y = ay;
        *(float2*)(aggr + (size_t)n * HID + lane * 2) = o;
    }
}

// ---------------------------------------------------------------------------
// x_out = relu( x_tag ⊗ W1l  +  aggr @ W2l  +  deg_attr ⊗ relu(W4l) )
// One wave computes a 16x64 tile via V_WMMA_F32_16X16X4_F32 (exact f32).
// W2l staged transposed in LDS with pad-68 stride (conflict-free b64 reads).
// ---------------------------------------------------------------------------
__global__ __launch_bounds__(256) void s2v_layer_kernel(
    const float* __restrict__ aggr, const float* __restrict__ xtag,
    const float* __restrict__ dega, const float* __restrict__ W2l,
    const float* __restrict__ W1l,  const float* __restrict__ W4l,
    float* __restrict__ xout, int nrows) {
    __shared__ float sB[64 * 68];  // sB[n*68 + k] = W2l[k][n]
    for (int i = threadIdx.x; i < 64 * 64; i += 256) {
        int k = i >> 6, n = i & 63;
        sB[n * 68 + k] = W2l[i];
    }
    __syncthreads();

    const int wave = threadIdx.x >> 5;
    const int lane = threadIdx.x & 31;
    const int base = (blockIdx.x * 8 + wave) * 16;
    if (base >= nrows) return;                 // wave-uniform: EXEC stays all-1s
    const int half = lane >> 4;
    const int m    = lane & 15;
    int arow = base + m; if (arow >= nrows) arow = nrows - 1;

    v8f c0 = {}, c1 = {}, c2 = {}, c3 = {};
    #pragma unroll
    for (int k0 = 0; k0 < 64; k0 += 4) {
        const int ka = k0 + 2 * half;          // A VGPR0/1 layout: K = ka, ka+1
        const float2 av = *(const float2*)(aggr + (size_t)arow * HID + ka);
        v2f a; a.x = av.x; a.y = av.y;
        v2f b0, b1, b2, b3;
        { const float2 t = *(const float2*)&sB[(0  + m) * 68 + ka]; b0.x = t.x; b0.y = t.y; }
        { const float2 t = *(const float2*)&sB[(16 + m) * 68 + ka]; b1.x = t.x; b1.y = t.y; }
        { const float2 t = *(const float2*)&sB[(32 + m) * 68 + ka]; b2.x = t.x; b2.y = t.y; }
        { const float2 t = *(const float2*)&sB[(48 + m) * 68 + ka]; b3.x = t.x; b3.y = t.y; }
        c0 = __builtin_amdgcn_wmma_f32_16x16x4_f32(false, a, false, b0, (short)0, c0, false, false);
        c1 = __builtin_amdgcn_wmma_f32_16x16x4_f32(false, a, false, b1, (short)0, c1, false, false);
        c2 = __builtin_amdgcn_wmma_f32_16x16x4_f32(false, a, false, b2, (short)0, c2, false, false);
        c3 = __builtin_amdgcn_wmma_f32_16x16x4_f32(false, a, false, b3, (short)0, c3, false, false);
    }

    // epilogue: D layout -> row = base + r + 8*half, col = nt*16 + m
    const float w1_0 = W1l[m],      w1_1 = W1l[16 + m];
    const float w1_2 = W1l[32 + m], w1_3 = W1l[48 + m];
    const float w4_0 = fmaxf(W4l[m], 0.f),      w4_1 = fmaxf(W4l[16 + m], 0.f);
    const float w4_2 = fmaxf(W4l[32 + m], 0.f), w4_3 = fmaxf(W4l[48 + m], 0.f);
    const int rbase = base + 8 * half;
    #pragma unroll
    for (int r = 0; r < 8; ++r) {
        const int orow = rbase + r;
        if (orow < nrows) {
            const float tg = xtag[orow];
            const float dg = dega[orow];
            float* o = xout + (size_t)orow * HID;
            o[m]      = fmaxf(c0[r] + tg * w1_0 + dg * w4_0, 0.f);
            o[16 + m] = fmaxf(c1[r] + tg * w1_1 + dg * w4_1, 0.f);
            o[32 + m] = fmaxf(c2[r] + tg * w1_2 + dg * w4_2, 0.f);
            o[48 + m] = fmaxf(c3[r] + tg * w1_3 + dg * w4_3, 0.f);
        }
    }
}

// ---------------------------------------------------------------------------
// sum_x[64] = column sum of x  (partial sums + f32 atomics)
// ---------------------------------------------------------------------------
__global__ void colsum_kernel(const float* __restrict__ x,
                              float* __restrict__ sumx, int nrows) {
    const int col    = threadIdx.x & 63;
    const int rstart = blockIdx.x * 4 + (threadIdx.x >> 6);
    const int rstep  = gridDim.x * 4;
    float acc = 0.f;
    for (int r = rstart; r < nrows; r += rstep) acc += x[(size_t)r * HID + col];
    atomicAdd(&sumx[col], acc);
}

// ---------------------------------------------------------------------------
// qg = sum_h relu( (sum_x @ W6)[h] ) * W5[h]     (single tiny block)
// ---------------------------------------------------------------------------
__global__ void qg_kernel(const float* __restrict__ sumx,
                          const float* __restrict__ W6,
                          const float* __restrict__ W5lo,
                          float* __restrict__ qg) {
    __shared__ float red[64];
    const int h = threadIdx.x;  // 0..63
    float acc = 0.f;
    for (int k = 0; k < 64; ++k) acc += sumx[k] * W6[k * 64 + h];
    red[h] = fmaxf(acc, 0.f) * W5lo[h];
    __syncthreads();
    if (h == 0) {
        float s = 0.f;
        for (int i = 0; i < 64; ++i) s += red[i];
        *qg = s;
    }
}

// ---------------------------------------------------------------------------
// Q[n] = qg + sum_h relu((x @ W7)[n,h]) * W5[64+h]
// Same WMMA tile as the layer kernel; shfl_xor row-reduction in epilogue.
// ---------------------------------------------------------------------------
__global__ __launch_bounds__(256) void q_kernel(
    const float* __restrict__ x, const float* __restrict__ W7,
    const float* __restrict__ W5hi, const float* __restrict__ qg,
    float* __restrict__ Q, int nrows) {
    __shared__ float sB[64 * 68];
    for (int i = threadIdx.x; i < 64 * 64; i += 256) {
        int k = i >> 6, n = i & 63;
        sB[n * 68 + k] = W7[i];
    }
    __syncthreads();

    const int wave = threadIdx.x >> 5;
    const int lane = threadIdx.x & 31;
    const int base = (blockIdx.x * 8 + wave) * 16;
    if (base >= nrows) return;
    const int half = lane >> 4;
    const int m    = lane & 15;
    int arow = base + m; if (arow >= nrows) arow = nrows - 1;

    v8f c0 = {}, c1 = {}, c2 = {}, c3 = {};
    #pragma unroll
    for (int k0 = 0; k0 < 64; k0 += 4) {
        const int ka = k0 + 2 * half;
        const float2 av = *(const float2*)(x + (size_t)arow * HID + ka);
        v2f a; a.x = av.x; a.y = av.y;
        v2f b0, b1, b2, b3;
        { const float2 t = *(const float2*)&sB[(0  + m) * 68 + ka]; b0.x = t.x; b0.y = t.y; }
        { const float2 t = *(const float2*)&sB[(16 + m) * 68 + ka]; b1.x = t.x; b1.y = t.y; }
        { const float2 t = *(const float2*)&sB[(32 + m) * 68 + ka]; b2.x = t.x; b2.y = t.y; }
        { const float2 t = *(const float2*)&sB[(48 + m) * 68 + ka]; b3.x = t.x; b3.y = t.y; }
        c0 = __builtin_amdgcn_wmma_f32_16x16x4_f32(false, a, false, b0, (short)0, c0, false, false);
        c1 = __builtin_amdgcn_wmma_f32_16x16x4_f32(false, a, false, b1, (short)0, c1, false, false);
        c2 = __builtin_amdgcn_wmma_f32_16x16x4_f32(false, a, false, b2, (short)0, c2, false, false);
        c3 = __builtin_amdgcn_wmma_f32_16x16x4_f32(false, a, false, b3, (short)0, c3, false, false);
    }

    const float w5_0 = W5hi[m],      w5_1 = W5hi[16 + m];
    const float w5_2 = W5hi[32 + m], w5_3 = W5hi[48 + m];
    const float qgv = *qg;
    const int rbase = base + 8 * half;
    #pragma unroll
    for (int r = 0; r < 8; ++r) {
        float acc = fmaxf(c0[r], 0.f) * w5_0 + fmaxf(c1[r], 0.f) * w5_1 +
                    fmaxf(c2[r], 0.f) * w5_2 + fmaxf(c3[r], 0.f) * w5_3;
        // reduce across the 16 lanes of this half (rows live per-half)
        acc += __shfl_xor(acc, 1);
        acc += __shfl_xor(acc, 2);
        acc += __shfl_xor(acc, 4);
        acc += __shfl_xor(acc, 8);
        if ((lane & 15) == 0) {
            const int orow = rbase + r;
            if (orow < nrows) Q[orow] = qgv + acc;
        }
    }
}

// ---------------------------------------------------------------------------
extern "C" void kernel_launch(void* const* d_in, const int* in_sizes, int n_in,
                              void* d_out, int out_size, void* d_ws, size_t ws_size,
                              hipStream_t stream) {
    const float* x0    = (const float*)d_in[0];
    const float* xtag  = (const float*)d_in[1];
    const float* eattr = (const float*)d_in[2];
    const int*   eidx  = (const int*)d_in[3];
    const float* W1    = (const float*)d_in[4];
    const float* W2    = (const float*)d_in[5];
    const float* W4    = (const float*)d_in[6];
    const float* W5    = (const float*)d_in[7];
    const float* W6    = (const float*)d_in[8];
    const float* W7    = (const float*)d_in[9];

    const int N = in_sizes[1];
    const int E = in_sizes[2];
    const int* src = eidx;
    const int* dst = eidx + E;

    // ---- workspace carve-up (256B aligned slices) ----
    char* ws = (char*)d_ws;
    size_t off = 0;
    auto alloc = [&](size_t bytes) {
        char* p = ws + off;
        off += (bytes + 255) & ~(size_t)255;
        return p;
    };
    const size_t nh = (size_t)N * HID * sizeof(float);
    float* bufA   = (float*)alloc(nh);
    float* bufB   = (float*)alloc(nh);
    float* aggr   = (float*)alloc(nh);
    float* dega   = (float*)alloc((size_t)N * sizeof(float));
    int*   hist   = (int*)  alloc((size_t)N * sizeof(int));
    int*   rowptr = (int*)  alloc((size_t)N * sizeof(int));
    int*   cursor = (int*)  alloc((size_t)N * sizeof(int));
    int*   colsrc = (int*)  alloc((size_t)E * sizeof(int));
    float* sumx   = (float*)alloc(64 * sizeof(float));
    float* qgb    = (float*)alloc(sizeof(float));

    // ---- build layer-invariant structures: deg_attr + dst-ordered CSR ----
    hipMemsetAsync(dega, 0, (size_t)N * sizeof(float), stream);
    hipMemsetAsync(hist, 0, (size_t)N * sizeof(int), stream);
    edge_stats_kernel<<<(E + 255) / 256, 256, 0, stream>>>(src, dst, eattr,
                                                           hist, dega, E);
    scan_kernel<<<1, 1024, 0, stream>>>(hist, rowptr, cursor, N);
    csr_fill_kernel<<<(E + 255) / 256, 256, 0, stream>>>(src, dst, cursor,
                                                         colsrc, E);

    // ---- T propagation rounds: atomic-free gather + WMMA GEMM ----
    const int gemmBlocks = (N + 127) / 128;  // 8 waves/block, 16 rows/wave
    const float* xcur = x0;
    for (int l = 0; l < T_LAYERS; ++l) {
        gather_aggr_kernel<<<2048, 256, 0, stream>>>(rowptr, hist, colsrc,
                                                     xcur, aggr, N);
        float* xnext = (l & 1) ? bufB : bufA;
        s2v_layer_kernel<<<gemmBlocks, 256, 0, stream>>>(
            aggr, xtag, dega, W2 + (size_t)l * HID * HID,
            W1 + (size_t)l * HID, W4 + (size_t)l * HID, xnext, N);
        xcur = xnext;
    }

    // ---- readout ----
    hipMemsetAsync(sumx, 0, 64 * sizeof(float), stream);
    colsum_kernel<<<256, 256, 0, stream>>>(xcur, sumx, N);
    qg_kernel<<<1, 64, 0, stream>>>(sumx, W6, W5, qgb);
    q_kernel<<<gemmBlocks, 256, 0, stream>>>(xcur, W7, W5 + HID, qgb,
                                             (float*)d_out, N);
}